// phase_modulation_29291676958774
// MI455X (gfx1250) — compile-verified
//
#include <hip/hip_runtime.h>

// out = (xr + i*xi) * exp(i * 2*pi * phase_padded)
// Shapes: x (bs=8, oc=3, ic=3, M=512, M=512) fp32; phase (3,3,256,256) fp32.
// Output complex64 -> interleaved float pairs, flat (bs,oc,ic,M,M,2).
//
// Pure bandwidth-bound streaming kernel:
//   - 4 complex elements per thread -> b128 loads/stores
//   - v_sin_f32 / v_cos_f32 are "revolution" units: sin(2*pi*x) == v_sin(x),
//     so __builtin_amdgcn_sinf(phase) computes exp's imaginary part directly.
//   - non-temporal hints on the 302MB streamed traffic; phase table (2.25MB,
//     reused 8x over batch) kept regular-temporal so it lives in L2.

typedef float __attribute__((ext_vector_type(4))) f4;

#define MDIM   512u
#define NDIM   256u
#define START  128u
#define CCOUNT 9u   // oc*ic

__global__ __launch_bounds__(256) void phase_mod_29291676958774_kernel(
    const float* __restrict__ xr,
    const float* __restrict__ xi,
    const float* __restrict__ phase,
    float* __restrict__ out,         // interleaved complex
    unsigned total_elems)            // bs*oc*ic*M*M (multiple of 4)
{
    unsigned t    = blockIdx.x * blockDim.x + threadIdx.x;
    unsigned base = t * 4u;                     // first complex element of this chunk
    if (base >= total_elems) return;

    // streamed, single-touch inputs -> non-temporal b128 loads
    f4 r = __builtin_nontemporal_load((const f4*)(xr + base));
    f4 q = __builtin_nontemporal_load((const f4*)(xi + base));

    // decompose flat index: (((b*9 + c)*512 + m)*512 + n)
    unsigned n   = base & (MDIM - 1u);
    unsigned m   = (base >> 9) & (MDIM - 1u);
    unsigned img = base >> 18;                  // b*9 + c
    unsigned c9  = img % CCOUNT;                // channel pair index (oc*3+ic)

    f4 o0, o1;

    // window test: wave-uniform (wave spans 128 consecutive n; bounds are
    // multiples of 128), so no divergence and out-of-window waves do zero
    // transcendental work.
    bool in_win = ((m - START) < NDIM) && ((n - START) < NDIM);
    if (in_win) {
        unsigned pidx = (c9 * NDIM + (m - START)) * NDIM + (n - START);
        f4 p = *(const f4*)(phase + pidx);      // RT load: keep table in L2 (8x reuse)
#pragma unroll
        for (int k = 0; k < 4; ++k) {
            // v_cos_f32 / v_sin_f32 compute cos/sin(2*pi * x) directly
            float c  = __builtin_amdgcn_cosf(p[k]);
            float s  = __builtin_amdgcn_sinf(p[k]);
            float re = r[k] * c - q[k] * s;
            float im = r[k] * s + q[k] * c;
            if (k < 2) { o0[2 * k]       = re; o0[2 * k + 1]       = im; }
            else       { o1[2 * (k - 2)] = re; o1[2 * (k - 2) + 1] = im; }
        }
    } else {
        // exp(i*2*pi*0) == 1 outside the window: pass-through (interleave only)
        o0[0] = r[0]; o0[1] = q[0]; o0[2] = r[1]; o0[3] = q[1];
        o1[0] = r[2]; o1[1] = q[2]; o1[2] = r[3]; o1[3] = q[3];
    }

    // interleaved complex output, 32B-aligned, streamed -> non-temporal b128 stores
    f4* dst = (f4*)(out + 2ull * (unsigned long long)base);
    __builtin_nontemporal_store(o0, dst);
    __builtin_nontemporal_store(o1, dst + 1);
}

extern "C" void kernel_launch(void* const* d_in, const int* in_sizes, int n_in,
                              void* d_out, int out_size, void* d_ws, size_t ws_size,
                              hipStream_t stream) {
    (void)n_in; (void)out_size; (void)d_ws; (void)ws_size;

    const float* xr    = (const float*)d_in[0];
    const float* xi    = (const float*)d_in[1];
    const float* phase = (const float*)d_in[2];
    float*       out   = (float*)d_out;

    unsigned total  = (unsigned)in_sizes[0];          // 8*3*3*512*512 = 18,874,368
    unsigned chunks = (total + 3u) / 4u;              // 4 complex elems per thread
    unsigned tpb    = 256u;                           // 8 wave32s per block
    unsigned blocks = (chunks + tpb - 1u) / tpb;      // 18,432 blocks

    phase_mod_29291676958774_kernel<<<blocks, tpb, 0, stream>>>(xr, xi, phase, out, total);
}